// NeRFPointSampler_73229192397439
// MI455X (gfx1250) — compile-verified
//
#include <hip/hip_runtime.h>
#include <stdint.h>

#define N_COARSE 64
#define N_IMP 128
#define N_ALL 192          // N_COARSE + N_IMP
#define RAYS_PER_BLOCK 8   // 8 waves of 32 => 256 threads
#define F_INF 3.402823466e+38f

// One wave (32 lanes, gfx1250 wave32) processes one ray.
__global__ __launch_bounds__(256) void nerf_fine_sample_kernel(
    const float* __restrict__ rays_o,
    const float* __restrict__ rays_d,
    const float* __restrict__ z_vals,
    const float* __restrict__ weights,
    float* __restrict__ pts_out,     // [N, 192, 3]
    float* __restrict__ zall_out,    // [N, 192]
    int n_rays)
{
    __shared__ __attribute__((aligned(16))) float s_zv [RAYS_PER_BLOCK][68];        // 64 + sentinel[64]
    __shared__ __attribute__((aligned(16))) float s_w  [RAYS_PER_BLOCK][N_COARSE];  // weights row
    __shared__ __attribute__((aligned(16))) float s_zm [RAYS_PER_BLOCK][N_COARSE];  // z_mid (63 used)
    __shared__ __attribute__((aligned(16))) float s_cdf[RAYS_PER_BLOCK][N_COARSE];  // cdf (63 used)
    __shared__ __attribute__((aligned(16))) float s_zs [RAYS_PER_BLOCK][132];       // 128 + sentinel[128]
    __shared__ __attribute__((aligned(16))) float s_za [RAYS_PER_BLOCK][N_ALL];     // merged z_all
    __shared__ __attribute__((aligned(16))) float s_pts[RAYS_PER_BLOCK][N_ALL * 3]; // pts row

    const int wave = threadIdx.x >> 5;
    const int lane = threadIdx.x & 31;
    const int ray  = blockIdx.x * RAYS_PER_BLOCK + wave;
    if (ray >= n_rays) return;   // wave-uniform predicate

    float* zv  = s_zv [wave];
    float* wb  = s_w  [wave];
    float* zm  = s_zm [wave];
    float* cdf = s_cdf[wave];
    float* zs  = s_zs [wave];
    float* za  = s_za [wave];
    float* pl  = s_pts[wave];

    // ---------------------------------------------------------------
    // Stage z_vals / weights rows into LDS via CDNA5 async global->LDS.
    // Each lane moves 8B: 32 lanes x 8B = one full 256B row per instr.
    // ---------------------------------------------------------------
    {
        const float* gz = z_vals  + (size_t)ray * N_COARSE + 2 * lane;
        const float* gw = weights + (size_t)ray * N_COARSE + 2 * lane;
        unsigned ldz = (unsigned)(uintptr_t)(&zv[2 * lane]); // low 32b of flat LDS ptr = LDS offset
        unsigned ldw = (unsigned)(uintptr_t)(&wb[2 * lane]);
        asm volatile("global_load_async_to_lds_b64 %0, %1, off"
                     :: "v"(ldz), "v"(gz) : "memory");
        asm volatile("global_load_async_to_lds_b64 %0, %1, off"
                     :: "v"(ldw), "v"(gw) : "memory");
        asm volatile("s_wait_asynccnt 0x0" ::: "memory");
    }
    if (lane == 0) zv[N_COARSE] = F_INF;   // merge sentinel

    // ---------------------------------------------------------------
    // z_mid[i] = 0.5*(z[i] + z[i+1]), i = 0..62   (2 indices per lane)
    // ---------------------------------------------------------------
    {
        const int i0 = lane;          // 0..31  (< 63 always)
        zm[i0] = 0.5f * (zv[i0] + zv[i0 + 1]);
        const int i1 = lane + 32;     // 32..63
        if (i1 < N_COARSE - 1)
            zm[i1] = 0.5f * (zv[i1] + zv[i1 + 1]);
    }

    // ---------------------------------------------------------------
    // w[i] = weights[i+1] + 1e-5, i = 0..61; CDF via wave32 shfl scan.
    // cdf[0] = 0, cdf[k] = prefix(w)[k-1] / sum(w), k = 1..62
    // ---------------------------------------------------------------
    {
        float a = wb[lane + 1] + 1e-5f;                         // i = lane (0..31, all < 62)
        float b = (lane < 30) ? (wb[lane + 33] + 1e-5f) : 0.0f; // i = lane+32 (32..61)

        float sl = a;
        #pragma unroll
        for (int d = 1; d < 32; d <<= 1) {
            float t = __shfl_up(sl, d, 32);
            if (lane >= d) sl += t;
        }
        float total_low = __shfl(sl, 31, 32);

        float sh = b;
        #pragma unroll
        for (int d = 1; d < 32; d <<= 1) {
            float t = __shfl_up(sh, d, 32);
            if (lane >= d) sh += t;
        }
        sh += total_low;
        float total = __shfl(sh, 31, 32);
        float inv = 1.0f / total;

        cdf[lane + 1] = sl * inv;                 // cdf[1..32]
        if (lane < 30) cdf[lane + 33] = sh * inv; // cdf[33..62]
        if (lane == 0) cdf[0] = 0.0f;
    }

    // ---------------------------------------------------------------
    // 128 inverse-CDF samples (4/lane): branchless searchsorted-right
    // over cdf[0..62] (6 fixed probes, step sum 63 == result range),
    // then linear interp between z_mid bins.
    // ---------------------------------------------------------------
    #pragma unroll
    for (int jj = 0; jj < 4; ++jj) {
        const int j = lane + jj * 32;
        const float u = (float)j * (1.0f / 127.0f);

        int lo = 0;                          // lo = #{k: cdf[k] <= u}, in [0,63]
        #pragma unroll
        for (int s = 32; s >= 1; s >>= 1)
            lo += (cdf[lo + s - 1] <= u) ? s : 0;

        const int below = (lo - 1 > 0) ? lo - 1 : 0;
        const int above = (lo < 62) ? lo : 62;

        const float cb = cdf[below], ca = cdf[above];
        const float bb = zm[below],  ba = zm[above];
        float denom = ca - cb;
        denom = (denom < 1e-5f) ? 1.0f : denom;
        const float t = (u - cb) / denom;
        zs[j] = fmaf(t, ba - bb, bb);
    }
    if (lane == 0) zs[N_IMP] = F_INF;      // merge sentinel

    // ---------------------------------------------------------------
    // Merge-path: lane owns merged outputs [6*lane, 6*lane+6).
    // Diagonal bisection finds a = #A-elems among first k merged elems
    // (A = zv, vals-first on ties; predicate zv[m] <= zs[k-m-1] is
    // monotone-decreasing in m). Then 6 sentinel-guarded merge steps,
    // computing pts directly from registers.
    // ---------------------------------------------------------------
    const float o0 = rays_o[(size_t)ray * 3 + 0];
    const float o1 = rays_o[(size_t)ray * 3 + 1];
    const float o2 = rays_o[(size_t)ray * 3 + 2];
    const float d0 = rays_d[(size_t)ray * 3 + 0];
    const float d1 = rays_d[(size_t)ray * 3 + 1];
    const float d2 = rays_d[(size_t)ray * 3 + 2];

    const int k = 6 * lane;                            // 0..186
    int alo = k - N_IMP;  alo = (alo > 0) ? alo : 0;   // max(0, k-128)
    int ahi = (k < N_COARSE) ? k : N_COARSE;           // min(64, k)

    int a = alo;
    #pragma unroll
    for (int s = 32; s >= 1; s >>= 1) {
        const int m  = a + s - 1;                      // may exceed range; masked below
        const int mc = (m < 63) ? m : 63;              // clamp for safe LDS read
        int bi = k - m - 1;
        bi = (bi > 0) ? bi : 0;                        // clamp; upper bound <=127 provable
        const int ok = (int)(m < ahi) & (int)(zv[mc] <= zs[bi]);
        a += ok ? s : 0;
    }
    {   // range-completing probe (result range [alo, ahi] has up to 65 values)
        const int ok = (a < ahi) && (zv[a] <= zs[k - a - 1]);  // a<ahi => indices in range
        a += ok ? 1 : 0;
    }
    int b = k - a;                                     // a in [alo,ahi], b in [0,128]

    float ca = zv[a];                                  // sentinel makes a==64 safe
    float cb = zs[b];                                  // sentinel makes b==128 safe
    float* plp = &pl[3 * k];
    #pragma unroll
    for (int e = 0; e < 6; ++e) {
        const int  takeA = (ca <= cb) ? 1 : 0;         // vals-first on ties
        const float z = fminf(ca, cb);
        za[k + e] = z;
        plp[3 * e + 0] = fmaf(d0, z, o0);
        plp[3 * e + 1] = fmaf(d1, z, o1);
        plp[3 * e + 2] = fmaf(d2, z, o2);
        a += takeA;                                    // a+b = k+e+1 <= 191 < 192:
        b += 1 - takeA;                                //   never both exhausted
        ca = zv[a];
        cb = zs[b];
    }

    // DS stores (DScnt) are unordered vs async (ASYNCcnt): drain LDS
    // writes before the async engine reads them.
    asm volatile("s_wait_dscnt 0x0" ::: "memory");

    // ---------------------------------------------------------------
    // Bulk async LDS->global stores: 4xB128 + 1xB64 for pts (2304B
    // row), 1xB128 + 1xB64 for z_all (768B row).
    // ---------------------------------------------------------------
    {
        float* pout = pts_out + (size_t)ray * (N_ALL * 3);
        #pragma unroll
        for (int t = 0; t < 4; ++t) {                 // 4 x 512B
            const float* g = pout + t * 128 + 4 * lane;
            unsigned l = (unsigned)(uintptr_t)(&pl[t * 128 + 4 * lane]);
            asm volatile("global_store_async_from_lds_b128 %0, %1, off"
                         :: "v"(g), "v"(l) : "memory");
        }
        {                                             // tail 256B
            const float* g = pout + 512 + 2 * lane;
            unsigned l = (unsigned)(uintptr_t)(&pl[512 + 2 * lane]);
            asm volatile("global_store_async_from_lds_b64 %0, %1, off"
                         :: "v"(g), "v"(l) : "memory");
        }
        float* zout = zall_out + (size_t)ray * N_ALL;
        {                                             // 512B
            const float* g = zout + 4 * lane;
            unsigned l = (unsigned)(uintptr_t)(&za[4 * lane]);
            asm volatile("global_store_async_from_lds_b128 %0, %1, off"
                         :: "v"(g), "v"(l) : "memory");
        }
        {                                             // tail 256B
            const float* g = zout + 128 + 2 * lane;
            unsigned l = (unsigned)(uintptr_t)(&za[128 + 2 * lane]);
            asm volatile("global_store_async_from_lds_b64 %0, %1, off"
                         :: "v"(g), "v"(l) : "memory");
        }
    }
    // S_ENDPGM performs an implicit wait-idle, draining ASYNCcnt.
}

extern "C" void kernel_launch(void* const* d_in, const int* in_sizes, int n_in,
                              void* d_out, int out_size, void* d_ws, size_t ws_size,
                              hipStream_t stream) {
    const float* rays_o  = (const float*)d_in[0];
    const float* rays_d  = (const float*)d_in[1];
    const float* z_vals  = (const float*)d_in[2];
    const float* weights = (const float*)d_in[3];

    const int n_rays = in_sizes[0] / 3;          // rays_o is [N,3]

    float* pts  = (float*)d_out;                                   // [N,192,3]
    float* zall = (float*)d_out + (size_t)n_rays * N_ALL * 3;      // [N,192]

    dim3 block(256);
    dim3 grid((n_rays + RAYS_PER_BLOCK - 1) / RAYS_PER_BLOCK);
    hipLaunchKernelGGL(nerf_fine_sample_kernel, grid, block, 0, stream,
                       rays_o, rays_d, z_vals, weights, pts, zall, n_rays);
}